// characteristic_16045997818331
// MI455X (gfx1250) — compile-verified
//
#include <hip/hip_runtime.h>
#include <cstdint>

// ---- problem constants (match reference) ----
#define ALPHA_F 0.939f
#define CFF_F   3.0f
#define EPS_F   1e-10f

constexpr int B_DIM   = 256;
constexpr int T_DIM   = 30000;
constexpr int C_DIM   = 3;

// ---- decomposition ----
constexpr int CHUNK   = 3008;                        // multiple of 32
constexpr int NCHUNK  = (T_DIM + CHUNK - 1) / CHUNK; // 10
constexpr int WARM    = 640;                         // multiple of 32; alpha^640 ~ 3e-18
constexpr int WPB     = 8;                           // waves per block (256 threads)
constexpr int NSTAGE  = 16;                          // LDS ring slots per wave
constexpr int PREF    = 14;                          // async copies kept in flight
constexpr int SLOT_F  = 32 * C_DIM;                  // floats per slot (96 = 384B)

static_assert(CHUNK % 32 == 0, "chunk must be multiple of 32");
static_assert(WARM % 32 == 0, "warm must be multiple of 32");
static_assert(PREF == 14, "s_wait_asynccnt immediate below assumes PREF==14");
static_assert(PREF + 2 <= NSTAGE, "need slack slots in LDS ring");
static_assert((NSTAGE & (NSTAGE - 1)) == 0, "power-of-two ring");
static_assert(WARM / 32 < (CHUNK / 32) - PREF, "warm phase must end before drain");

// DPP row_shr:n within 16-lane rows, zero-fill for invalid source lanes
#define DPP_ROW_SHR(x, n)                                                   \
    __int_as_float(__builtin_amdgcn_update_dpp(                             \
        0, __float_as_int(x), 0x110 | (n), 0xF, 0xF, true))

#define READLANE_F(x, l)                                                    \
    __int_as_float(__builtin_amdgcn_readlane(__float_as_int(x), (l)))

__global__ __launch_bounds__(WPB * 32)
void iir_charfunc_kernel(const float* __restrict__ rdata, float* __restrict__ out)
{
    __shared__ float sbuf[WPB * NSTAGE * SLOT_F];   // 48 KB / block

    const int lane = threadIdx.x & 31;
    const int wib  = threadIdx.x >> 5;
    const int wave = blockIdx.x * WPB + wib;
    const int b    = wave / NCHUNK;
    const int ck   = wave - b * NCHUNK;
    if (b >= B_DIM) return;

    const int t0     = ck * CHUNK;
    const int tendC  = min(T_DIM, t0 + CHUNK);
    const int tstart = max(0, t0 - WARM);
    const int nG     = (tendC - tstart + 31) >> 5;   // >= 94 groups of 32 steps
    const int gWarm  = (t0 - tstart) >> 5;           // 0 or 20 warm-up groups
    const int gMain  = nG - PREF;                    // drain boundary (>= 80)

    const float* __restrict__ gbase = rdata + (size_t)b * (T_DIM * C_DIM);
    float*       __restrict__ obase = out   + (size_t)b * (T_DIM * C_DIM);
    float* myLds = sbuf + wib * (NSTAGE * SLOT_F);
    const uint32_t ldsBase = (uint32_t)(uintptr_t)myLds;  // low 32b = LDS offset

    // issue one 384B group tile: 24 lanes x b128 async copies into ring slot
    auto issue = [&](int gi) {
        const int tg    = tstart + (gi << 5);
        const int quads = min(24, ((T_DIM - tg) * 3) >> 2);  // 16B quads (tail-safe)
        if (lane < quads) {
            const uint32_t la = ldsBase
                              + (uint32_t)((gi & (NSTAGE - 1)) * SLOT_F * 4)
                              + (uint32_t)(lane * 16);
            const char* ga = (const char*)gbase + (size_t)tg * (C_DIM * 4)
                           + (size_t)lane * 16;
            asm volatile("global_load_async_to_lds_b128 %0, %1, off"
                         :: "v"(la), "v"(ga) : "memory");
        }
    };

    // start the async pipeline immediately
    for (int i = 0; i < PREF; ++i) issue(i);

    // boundary element r[tstart-1] (or 0) -> last element of ring slot 15,
    // so the g==0 neighbor read works like every other iteration.
    if (lane == 0) {
        float pl0 = 0.f, pl1 = 0.f, pl2 = 0.f;
        if (tstart > 0) {
            const float* pp = gbase + (size_t)(tstart - 1) * C_DIM;
            pl0 = pp[0]; pl1 = pp[1]; pl2 = pp[2];
        }
        float* q = myLds + (NSTAGE - 1) * SLOT_F + 31 * C_DIM;
        q[0] = pl0; q[1] = pl1; q[2] = pl2;
    }

    // constant-folded alpha powers
    const float a1 = ALPHA_F;
    const float a2 = a1 * a1, a4 = a2 * a2, a8 = a4 * a4, a16 = a8 * a8, a32 = a16 * a16;
    const float INVA = 1.0f / ALPHA_F;      // diff = K1*y + INVA*x
    const float K1   = 1.0f - INVA;

    // pa = alpha^(lane+1)  (carry-in weight)
    int e = lane + 1;
    float pa = 1.0f;
    if (e & 1)  pa *= a1;
    if (e & 2)  pa *= a2;
    if (e & 4)  pa *= a4;
    if (e & 8)  pa *= a8;
    if (e & 16) pa *= a16;
    if (e & 32) pa *= a32;

    // mrow = (lane>=16) ? alpha^(lane-15) : 0   (cross-row scan fix-up weight)
    float mrow = 0.0f;
    if (lane >= 16) {
        int e2 = lane - 15;
        mrow = 1.0f;
        if (e2 & 1)  mrow *= a1;
        if (e2 & 2)  mrow *= a2;
        if (e2 & 4)  mrow *= a4;
        if (e2 & 8)  mrow *= a8;
        if (e2 & 16) mrow *= a16;
    }

    float c0 = 0.f, c1 = 0.f, c2 = 0.f;     // carry: d state at (t-1), wave-uniform

    // STORE_MODE: 0 = warm-up (no stores), 1 = unguarded stores, 2 = tail-guarded
    auto body = [&](int g, int STORE_MODE) {
        const int slot = g & (NSTAGE - 1);

        const float* sp = myLds + slot * SLOT_F + lane * C_DIM;
        float r0 = sp[0], r1 = sp[1], r2 = sp[2];

        // previous element: last 12B of previous ring slot for lane 0
        const int pslot = (g - 1) & (NSTAGE - 1);
        const float* pp = (lane == 0)
                        ? (myLds + pslot * SLOT_F + 31 * C_DIM)
                        : (sp - C_DIM);
        float x0 = r0 - pp[0] + EPS_F;
        float x1 = r1 - pp[1] + EPS_F;
        float x2 = r2 - pp[2] + EPS_F;

        // decayed inclusive scan within 16-lane rows (DPP, zero-fill)
        float v0 = x0, v1 = x1, v2 = x2;
        v0 = fmaf(a1, DPP_ROW_SHR(v0, 1), v0);
        v1 = fmaf(a1, DPP_ROW_SHR(v1, 1), v1);
        v2 = fmaf(a1, DPP_ROW_SHR(v2, 1), v2);
        v0 = fmaf(a2, DPP_ROW_SHR(v0, 2), v0);
        v1 = fmaf(a2, DPP_ROW_SHR(v1, 2), v1);
        v2 = fmaf(a2, DPP_ROW_SHR(v2, 2), v2);
        v0 = fmaf(a4, DPP_ROW_SHR(v0, 4), v0);
        v1 = fmaf(a4, DPP_ROW_SHR(v1, 4), v1);
        v2 = fmaf(a4, DPP_ROW_SHR(v2, 4), v2);
        v0 = fmaf(a8, DPP_ROW_SHR(v0, 8), v0);
        v1 = fmaf(a8, DPP_ROW_SHR(v1, 8), v1);
        v2 = fmaf(a8, DPP_ROW_SHR(v2, 8), v2);
        // cross-row: lanes 16..31 add alpha^(lane-15) * scan(lane 15)
        v0 = fmaf(mrow, READLANE_F(v0, 15), v0);
        v1 = fmaf(mrow, READLANE_F(v1, 15), v1);
        v2 = fmaf(mrow, READLANE_F(v2, 15), v2);

        // apply carry-in: y = v + alpha^(lane+1) * d_{-1}
        float y0 = fmaf(pa, c0, v0);
        float y1 = fmaf(pa, c1, v1);
        float y2 = fmaf(pa, c2, v2);

        if (STORE_MODE != 0) {
            const int t = tstart + (g << 5) + lane;
            if (STORE_MODE == 1 || t < tendC) {
                // diff = y - y_prev = (1-1/a)*y + (1/a)*x  (from the recurrence)
                float d0 = fmaf(K1, y0, INVA * x0);
                float d1 = fmaf(K1, y1, INVA * x1);
                float d2 = fmaf(K1, y2, INVA * x2);
                float* op = obase + (size_t)t * C_DIM;
                op[0] = fmaf(y0, y0, CFF_F * (d0 * d0));
                op[1] = fmaf(y1, y1, CFF_F * (d1 * d1));
                op[2] = fmaf(y2, y2, CFF_F * (d2 * d2));
            }
        }

        // carry for next group = y at lane 31 (wave-uniform scalar)
        c0 = READLANE_F(y0, 31);
        c1 = READLANE_F(y1, 31);
        c2 = READLANE_F(y2, 31);
    };

    int g = 0;
    // Phase A: warm-up, no stores
    for (; g < gWarm; ++g) {
        asm volatile("s_wait_asynccnt 13" ::: "memory");   // PREF-1 still in flight
        body(g, 0);
        issue(g + PREF);
    }
    // Phase B: steady state, unguarded stores
    for (; g < gMain; ++g) {
        asm volatile("s_wait_asynccnt 13" ::: "memory");
        body(g, 1);
        issue(g + PREF);
    }
    // Phase C: drain tail, guarded stores
    for (; g < nG; ++g) {
        asm volatile("s_wait_asynccnt 0" ::: "memory");
        body(g, 2);
    }
}

extern "C" void kernel_launch(void* const* d_in, const int* in_sizes, int n_in,
                              void* d_out, int out_size, void* d_ws, size_t ws_size,
                              hipStream_t stream) {
    (void)in_sizes; (void)n_in; (void)out_size; (void)d_ws; (void)ws_size;
    const float* rdata = (const float*)d_in[0];
    float* out = (float*)d_out;
    const int totalWaves = B_DIM * NCHUNK;            // 2560
    const int nBlocks = (totalWaves + WPB - 1) / WPB; // 320
    iir_charfunc_kernel<<<nBlocks, WPB * 32, 0, stream>>>(rdata, out);
}